// CNNS4D_6356551598499
// MI455X (gfx1250) — compile-verified
//
#include <hip/hip_runtime.h>
#include <hip/hip_bf16.h>
#include <math.h>

// ---------------------------------------------------------------------------
// CNN + S4D forward for MI455X (gfx1250), fp32 WMMA path.
// Convs = implicit GEMM on v_wmma_f32_16x16x4_f32 (full fp32 precision,
// matching the fp32 reference). Weight tiles are staged into LDS with
// gfx1250 async global->LDS loads (ASYNCcnt) when available.
// Downstream of the convs only the last timestep survives, so the FFT conv
// collapses to one causal dot product per (b,h).
// ---------------------------------------------------------------------------

typedef __attribute__((ext_vector_type(2))) float v2f;
typedef __attribute__((ext_vector_type(8))) float v8f;

#define CONV_KB 16    // K chunk staged in LDS (4 wmma k-steps)
#define CONV_BM 128   // M tile per block
#define CONV_BN 64    // N tile per block

#if __has_builtin(__builtin_amdgcn_global_load_async_to_lds_b32) && \
    __has_builtin(__builtin_amdgcn_s_wait_asynccnt)
#define USE_ASYNC_LDS 1
// builtin signature (from diagnostics): (global int*, local int*, imm offset, imm cpol)
#define ASYNC_G2L_B32(gp, lp)                                          \
    __builtin_amdgcn_global_load_async_to_lds_b32(                     \
        (__attribute__((address_space(1))) int*)(gp),                  \
        (__attribute__((address_space(3))) int*)(lp), 0, 0)
#else
#define USE_ASYNC_LDS 0
#endif

// ---------------------------------------------------------------------------
// Generic 3x3 stride-2 pad-1 conv + per-channel scale/bias + ReLU.
// out[img][n][oy][ox] = relu( scale[n] * sum_k A[row][k]*W[n][k] + bias[n] )
// A = im2col(in), row = (img,oy,ox), k = ic*9 + kh*3 + kw, W row-major [Cout][K].
// Block: 256 threads = 8 waves, tile 128x64. Wave (wm,wn) owns a 32x32 slab:
// 2 A-fragments x 2 B-fragments -> 4 fp32 WMMA accumulators, 16 WMMAs/chunk.
// OH == OW are powers of two for all layers -> shifts instead of divides.
// ---------------------------------------------------------------------------
__global__ __launch_bounds__(256) void conv_wmma_kernel(
    const float* __restrict__ in, const float* __restrict__ w,
    const float* __restrict__ gain, const float* __restrict__ bias,
    float* __restrict__ out,
    int Nimg, int Cin, int Hin, int Win, int Cout, int owShift)
{
    const int OW = 1 << owShift, OH = OW;
    const int pixShift = 2 * owShift;          // log2(OH*OW)
    const int pixMask  = (1 << pixShift) - 1;
    const int owMask   = OW - 1;
    const int K  = Cin * 9;
    const int M  = Nimg << pixShift;
    const int tileM = blockIdx.x * CONV_BM;
    const int tileN = blockIdx.y * CONV_BN;

    __shared__ float As[CONV_BM][CONV_KB + 2]; // +2 pad: stride 18 -> conflict-free frag reads
    __shared__ float Bs[CONV_KB][CONV_BN];

    const int tid  = threadIdx.x;
    const int lane = tid & 31;
    const int wv   = tid >> 5;     // wave 0..7
    const int wm   = wv >> 1;      // 0..3  (M slab: rows wm*32..+31)
    const int wn   = wv & 1;       // 0..1  (N slab: cols wn*32..+31)
    const int half = lane >> 4;    // hi/lo half-wave
    const int l16  = lane & 15;

    v8f acc[2][2];
    #pragma unroll
    for (int i = 0; i < 2; ++i)
        #pragma unroll
        for (int j = 0; j < 2; ++j)
            acc[i][j] = (v8f){0.f, 0.f, 0.f, 0.f, 0.f, 0.f, 0.f, 0.f};

    const bool nFull = (tileN + CONV_BN) <= Cout;

    for (int k0 = 0; k0 < K; k0 += CONV_KB) {
        // ---- stage A tile 128x16 (im2col gather, zero-padded) ----
        #pragma unroll
        for (int it = 0; it < (CONV_BM * CONV_KB) / 256; ++it) {
            const int idx = tid + it * 256;
            const int r  = idx >> 4;       // row within tile (KB==16)
            const int kk = idx & 15;
            const int k  = k0 + kk;
            float v = 0.f;
            const int row = tileM + r;
            if (k < K && row < M) {
                const int img = row >> pixShift;
                const int rem = row & pixMask;
                const int oy  = rem >> owShift;
                const int ox  = rem & owMask;
                const int ic  = k / 9;          // constant divisor -> mul/shift
                const int r9  = k - ic * 9;
                const int kh  = r9 / 3;
                const int kw  = r9 - kh * 3;
                const int iy  = oy * 2 - 1 + kh;
                const int ix  = ox * 2 - 1 + kw;
                if (iy >= 0 && iy < Hin && ix >= 0 && ix < Win)
                    v = in[((size_t)(img * Cin + ic) * Hin + iy) * Win + ix];
            }
            As[r][kk] = v;
        }
        // ---- stage B tile 16x64 (weights, transposed gather) ----
        if (nFull && (k0 + CONV_KB) <= K) {
#if USE_ASYNC_LDS
            // gfx1250 async global->LDS copy, one b32 per lane per issue
            #pragma unroll
            for (int j = 0; j < (CONV_KB * CONV_BN) / 256; ++j) {
                const int idx = tid + j * 256;
                const int k = idx & 15;        // lanes sweep k -> coalesced 64B runs
                const int n = idx >> 4;
                ASYNC_G2L_B32(&w[(size_t)(tileN + n) * K + (k0 + k)], &Bs[k][n]);
            }
#else
            #pragma unroll
            for (int j = 0; j < (CONV_KB * CONV_BN) / 256; ++j) {
                const int idx = tid + j * 256;
                const int k = idx & 15;
                const int n = idx >> 4;
                Bs[k][n] = w[(size_t)(tileN + n) * K + (k0 + k)];
            }
#endif
        } else {
            // edge path (layer 1: K=27, Cout=32): guarded + zero fill
            #pragma unroll
            for (int j = 0; j < (CONV_KB * CONV_BN) / 256; ++j) {
                const int idx = tid + j * 256;
                const int k = idx & 15;
                const int n = idx >> 4;
                const int gk = k0 + k;
                const int gn = tileN + n;
                float v = 0.f;
                if (gk < K && gn < Cout) v = w[(size_t)gn * K + gk];
                Bs[k][n] = v;
            }
            if (k0 + CONV_KB < K && (tileN + (tid & 63)) < Cout)
                __builtin_prefetch(&w[(size_t)(tileN + (tid & 63)) * K + k0 + CONV_KB], 0, 0);
        }
#if USE_ASYNC_LDS
        __builtin_amdgcn_s_wait_asynccnt(0);
#endif
        __syncthreads();

        // ---- 4 WMMA k-steps, 4 accumulators each ----
        #pragma unroll
        for (int ks = 0; ks < CONV_KB; ks += 4) {
            const int kb = ks + 2 * half;   // lanes 0-15: K {ks,ks+1}; 16-31: {ks+2,ks+3}
            v2f a[2], b[2];
            a[0] = *(const v2f*)&As[wm * 32 + l16][kb];
            a[1] = *(const v2f*)&As[wm * 32 + 16 + l16][kb];
            b[0].x = Bs[kb][wn * 32 + l16];
            b[0].y = Bs[kb + 1][wn * 32 + l16];
            b[1].x = Bs[kb][wn * 32 + 16 + l16];
            b[1].y = Bs[kb + 1][wn * 32 + 16 + l16];
            #pragma unroll
            for (int i = 0; i < 2; ++i)
                #pragma unroll
                for (int j = 0; j < 2; ++j)
                    acc[i][j] = __builtin_amdgcn_wmma_f32_16x16x4_f32(
                        false, a[i], false, b[j], (short)0, acc[i][j], false, false);
        }
        __syncthreads();
    }

    // ---- fused scale/bias/ReLU epilogue, NCHW scatter ----
    const float rs = rsqrtf(1.0f + 1e-5f);
    #pragma unroll
    for (int i = 0; i < 2; ++i) {
        #pragma unroll
        for (int v = 0; v < 8; ++v) {
            const int m   = wm * 32 + i * 16 + v + half * 8; // C/D: VGPR v -> M=v(+8)
            const int row = tileM + m;
            if (row >= M) continue;
            const int img = row >> pixShift;
            const int rem = row & pixMask;
            const int oy  = rem >> owShift;
            const int ox  = rem & owMask;
            #pragma unroll
            for (int j = 0; j < 2; ++j) {
                const int n = tileN + wn * 32 + j * 16 + l16;
                if (n < Cout) {
                    float val = acc[i][j][v];
                    val = val * (gain[n] * rs) + bias[n];
                    val = val > 0.f ? val : 0.f;
                    out[((size_t)(img * Cout + n) * OH + oy) * OW + ox] = val;
                }
            }
        }
    }
}

// ---------------------------------------------------------------------------
// 4x4 global mean pool + (B,T,DM) -> (B,DM,T) transpose.
// in : (512, 1280, 4, 4)   u : u[(b*1280+d)*128 + t]
// ---------------------------------------------------------------------------
__global__ void pool_kernel(const float* __restrict__ in, float* __restrict__ u)
{
    const int idx = blockIdx.x * blockDim.x + threadIdx.x;  // over 512*1280
    if (idx >= 512 * 1280) return;
    const int i = idx / 1280;            // image = b*128 + t
    const int d = idx - i * 1280;
    const float* p = in + (size_t)idx * 16;
    float s = 0.f;
    #pragma unroll
    for (int q = 0; q < 16; ++q) s += p[q];
    s *= (1.0f / 16.0f);
    const int b = i >> 7;
    const int t = i & 127;
    u[((size_t)b * 1280 + d) * 128 + t] = s;
}

// ---------------------------------------------------------------------------
// S4D kernel:  k[h][m] = 2 * sum_n Ct[h][n] * exp(dtA[h][n] * m)
// where dtA = A*dt, A = -exp(log_A_real), dt = exp(log_dt),
//       Ct = C * (exp(dtA)-1)/A.   One block per h, thread = m.
// ---------------------------------------------------------------------------
__global__ __launch_bounds__(128) void kfilter_kernel(
    const float* __restrict__ log_dt, const float* __restrict__ C,
    const float* __restrict__ log_A_real, float* __restrict__ kout)
{
    const int h = blockIdx.x;     // 0..1279
    const int m = threadIdx.x;    // 0..127
    __shared__ float sCt[32], sdtA[32];
    if (m < 32) {
        const float dt  = expf(log_dt[h]);
        const float A   = -expf(log_A_real[h * 32 + m]);
        const float dtA = A * dt;
        sdtA[m] = dtA;
        sCt[m]  = C[h * 32 + m] * (expf(dtA) - 1.0f) / A;
    }
    __syncthreads();
    float s = 0.f;
    #pragma unroll 4
    for (int n = 0; n < 32; ++n) s += sCt[n] * expf(sdtA[n] * (float)m);
    kout[(size_t)h * 128 + m] = 2.0f * s;
}

// ---------------------------------------------------------------------------
// Last-timestep causal conv + skip + exact GELU:
// yl[b][h] = gelu( sum_m k[h][m]*u[b][h][127-m] + D[h]*u[b][h][127] )
// ---------------------------------------------------------------------------
__global__ void ylast_kernel(const float* __restrict__ u, const float* __restrict__ kf,
                             const float* __restrict__ D, float* __restrict__ yl)
{
    const int idx = blockIdx.x * blockDim.x + threadIdx.x;  // 4*1280
    if (idx >= 4 * 1280) return;
    const int b = idx / 1280;
    const int h = idx - b * 1280;
    const float* up = u + (size_t)idx * 128;
    const float* kp = kf + (size_t)h * 128;
    float s = 0.f;
    #pragma unroll 4
    for (int m = 0; m < 128; ++m) s += kp[m] * up[127 - m];
    s += D[h] * up[127];
    yl[idx] = 0.5f * s * (1.0f + erff(s * 0.70710678118654752f));
}

// ---------------------------------------------------------------------------
// z[b][o] = ob[o] + ow[o,:] . yl[b,:]   (one wave32 per output, lane reduce)
// ---------------------------------------------------------------------------
__global__ __launch_bounds__(256) void zproj_kernel(
    const float* __restrict__ ow, const float* __restrict__ ob,
    const float* __restrict__ yl, float* __restrict__ z)
{
    const int wvg  = (blockIdx.x * blockDim.x + threadIdx.x) >> 5;
    const int lane = threadIdx.x & 31;
    if (wvg >= 4 * 2560) return;
    const int b = wvg / 2560;
    const int o = wvg - b * 2560;
    const float* wr = ow + (size_t)o * 1280;
    const float* yp = yl + (size_t)b * 1280;
    float s = 0.f;
    for (int h = lane; h < 1280; h += 32) s += wr[h] * yp[h];
    #pragma unroll
    for (int off = 16; off > 0; off >>= 1) s += __shfl_down(s, off, 32);
    if (lane == 0) z[(size_t)b * 2560 + o] = s + ob[o];
}

// glu[b][h] = z[b][h] * sigmoid(z[b][1280+h])
__global__ void glu_kernel(const float* __restrict__ z, float* __restrict__ gl)
{
    const int idx = blockIdx.x * blockDim.x + threadIdx.x;  // 4*1280
    if (idx >= 4 * 1280) return;
    const int b = idx / 1280;
    const int h = idx - b * 1280;
    const float a = z[(size_t)b * 2560 + h];
    const float g = z[(size_t)b * 2560 + 1280 + h];
    gl[idx] = a * (1.0f / (1.0f + expf(-g)));
}

// hid = relu(gl @ rw1^T + rb1);  out = hid @ rw2^T + rb2   (block per batch)
__global__ __launch_bounds__(64) void head_kernel(
    const float* __restrict__ gl, const float* __restrict__ rw1,
    const float* __restrict__ rb1, const float* __restrict__ rw2,
    const float* __restrict__ rb2, float* __restrict__ out)
{
    const int b = blockIdx.x;   // 0..3
    const int j = threadIdx.x;  // 0..63
    __shared__ float hid[64];
    const float* gp = gl + (size_t)b * 1280;
    float s = rb1[j];
    for (int h = 0; h < 1280; ++h) s += rw1[(size_t)j * 1280 + h] * gp[h];
    hid[j] = s > 0.f ? s : 0.f;
    __syncthreads();
    if (j < 60) {
        float o = rb2[j];
        #pragma unroll
        for (int q = 0; q < 64; ++q) o += rw2[(size_t)j * 64 + q] * hid[q];
        out[(size_t)b * 60 + j] = o;
    }
}

// ---------------------------------------------------------------------------
extern "C" void kernel_launch(void* const* d_in, const int* in_sizes, int n_in,
                              void* d_out, int out_size, void* d_ws, size_t ws_size,
                              hipStream_t stream)
{
    const float* x   = (const float*)d_in[0];
    const float* cw[5] = { (const float*)d_in[1],  (const float*)d_in[4],
                           (const float*)d_in[7],  (const float*)d_in[10],
                           (const float*)d_in[13] };
    const float* gg[5] = { (const float*)d_in[2],  (const float*)d_in[5],
                           (const float*)d_in[8],  (const float*)d_in[11],
                           (const float*)d_in[14] };
    const float* bb[5] = { (const float*)d_in[3],  (const float*)d_in[6],
                           (const float*)d_in[9],  (const float*)d_in[12],
                           (const float*)d_in[15] };
    const float* log_dt     = (const float*)d_in[16];
    const float* Cm         = (const float*)d_in[17];
    const float* log_A_real = (const float*)d_in[18];
    const float* Dv         = (const float*)d_in[19];
    const float* ow         = (const float*)d_in[20];
    const float* ob         = (const float*)d_in[21];
    const float* rw1        = (const float*)d_in[22];
    const float* rb1        = (const float*)d_in[23];
    const float* rw2        = (const float*)d_in[24];
    const float* rb2        = (const float*)d_in[25];
    float* outp = (float*)d_out;

    char* ws = (char*)d_ws;
    float* A0  = (float*)(ws);                          // 268,435,456 B (max act)
    float* A1  = (float*)(ws + 268435456ull);           // 134,217,728 B
    float* U   = (float*)(ws + 402653184ull);           // 2,621,440 B
    float* Kf  = (float*)(ws + 405274624ull);           //   655,360 B
    float* YL  = (float*)(ws + 405929984ull);           //    20,480 B
    float* Zb  = (float*)(ws + 405950464ull);           //    40,960 B
    float* GL  = (float*)(ws + 405991424ull);           //    20,480 B

    // ---- conv stack (all via fp32 WMMA implicit GEMM) ----
    struct LayerDesc { int Cin, Hin, Cout; };
    const LayerDesc L[5] = { {3,128,32}, {32,64,64}, {64,32,128}, {128,16,512}, {512,8,1280} };
    const float* src = x;
    float* bufs[2] = { A0, A1 };
    for (int l = 0; l < 5; ++l) {
        const int Cin = L[l].Cin, Hin = L[l].Hin, Cout = L[l].Cout;
        const int OH = Hin >> 1;
        const int owShift = __builtin_ctz(OH);
        const int Mtot = 512 * OH * OH;
        float* dst = bufs[l & 1];
        dim3 grid((Mtot + CONV_BM - 1) / CONV_BM, (Cout + CONV_BN - 1) / CONV_BN);
        conv_wmma_kernel<<<grid, 256, 0, stream>>>(
            src, cw[l], gg[l], bb[l], dst, 512, Cin, Hin, Hin, Cout, owShift);
        src = dst;
    }
    const float* conv_out = src;   // A0: (512, 1280, 4, 4)

    // ---- pool + transpose ----
    pool_kernel<<<(512 * 1280 + 255) / 256, 256, 0, stream>>>(conv_out, U);

    // ---- S4D kernel k[h][m] ----
    kfilter_kernel<<<1280, 128, 0, stream>>>(log_dt, Cm, log_A_real, Kf);

    // ---- last-timestep causal conv + GELU ----
    ylast_kernel<<<(4 * 1280 + 255) / 256, 256, 0, stream>>>(U, Kf, Dv, YL);

    // ---- output projection (wave per output) + GLU ----
    zproj_kernel<<<(4 * 2560 * 32 + 255) / 256, 256, 0, stream>>>(ow, ob, YL, Zb);
    glu_kernel<<<(4 * 1280 + 255) / 256, 256, 0, stream>>>(Zb, GL);

    // ---- MLP head -> (4, 60) ----
    head_kernel<<<4, 64, 0, stream>>>(GL, rw1, rb1, rw2, rb2, outp);
}